// LayerSelectorMoE_33543694581813
// MI455X (gfx1250) — compile-verified
//
#include <hip/hip_runtime.h>
#include <hip/hip_bf16.h>

#define B_    8
#define LP1   25
#define L_    24
#define T_    512
#define D_    1024
#define P_    128
#define TCH1  8      // T chunks for score kernel (64 rows each)
#define TROW1 64
#define TCH3  16     // T chunks for gather/max kernel (32 rows each)
#define TROW3 32

// workspace layout (float offsets)
#define OFF_SCORE 0                          // 192*TCH1 = 1536 floats: score partial sums
#define OFF_TOPV  1536                       // 24 floats: top-3 softmax weights
#define OFF_TOPI  1600                       // 24 ints:   top-3 layer indices
#define OFF_PMAX  2048                       // B*TCH3*D = 131072 floats: partial maxima
#define OFF_NORM  (2048 + B_*TCH3*D_)        // 16*1024 floats: zero-padded normed
// total ~ 600 KB of ws

// output layout (floats): projected[1024] | scores[192] | topk_idx[24] | selected[8*3*512*1024]
#define OUT_SCORES 1024
#define OUT_TOPI   (1024 + 192)
#define OUT_SEL    (1024 + 192 + 24)

typedef float v2f __attribute__((ext_vector_type(2)));
typedef float v8f __attribute__((ext_vector_type(8)));

__global__ void k0_init(float* __restrict__ ws) {
    int i = blockIdx.x * 256 + threadIdx.x;
    if (i < 8 * D_) ws[OFF_NORM + 8 * D_ + i] = 0.0f;   // zero pad rows 8..15 of normed
}

// score partial dot: one block per (b,l,t-chunk); thread tid owns d = 4*tid..4*tid+3
// writes a per-chunk partial sum (no atomics -> bitwise deterministic)
__global__ void k1_scores(const float* __restrict__ x, const float* __restrict__ wsc,
                          float* __restrict__ ws) {
    __shared__ float red[256];
    const int tid = threadIdx.x;
    const int bl  = blockIdx.x;          // 0..191
    const int b   = bl / L_, l = bl % L_;
    const int tc  = blockIdx.y;
    const float4 w4 = ((const float4*)wsc)[tid];
    const float4* x4 = (const float4*)(x + ((size_t)(b * LP1 + l + 1) * T_ + (size_t)tc * TROW1) * D_);
    float acc = 0.0f;
    #pragma unroll 4
    for (int tt = 0; tt < TROW1; ++tt) {
        float4 v = x4[tt * (D_ / 4) + tid];
        acc += v.x * w4.x + v.y * w4.y + v.z * w4.z + v.w * w4.w;
    }
    red[tid] = acc;
    __syncthreads();
    for (int s = 128; s > 0; s >>= 1) {
        if (tid < s) red[tid] += red[tid + s];
        __syncthreads();
    }
    if (tid == 0) ws[OFF_SCORE + bl * TCH1 + tc] = red[0];
}

// reduce chunk partials, softmax + top-3 per batch (one thread per batch; tiny)
__global__ void k2_topk(const float* __restrict__ bsc, float* __restrict__ ws,
                        float* __restrict__ out) {
    int b = threadIdx.x;
    if (b >= B_) return;
    float s[L_];
    float mx = -3.402823466e38f;
    for (int l = 0; l < L_; ++l) {
        float raw = 0.0f;
        for (int c = 0; c < TCH1; ++c) raw += ws[OFF_SCORE + (b * L_ + l) * TCH1 + c];
        float v = raw * (1.0f / T_) + bsc[0];
        s[l] = v;
        out[OUT_SCORES + b * L_ + l] = v;
        mx = fmaxf(mx, v);
    }
    float p[L_], sum = 0.0f;
    for (int l = 0; l < L_; ++l) { p[l] = __expf(s[l] - mx); sum += p[l]; }
    float inv = 1.0f / sum;
    for (int l = 0; l < L_; ++l) p[l] *= inv;
    int* wsi = (int*)ws;
    for (int j = 0; j < 3; ++j) {
        int bi = 0; float bv = -1.0f;
        for (int l = 0; l < L_; ++l) if (p[l] > bv) { bv = p[l]; bi = l; }
        p[bi] = -2.0f;
        ws[OFF_TOPV + b * 3 + j] = bv;
        wsi[OFF_TOPI + b * 3 + j] = bi;
        out[OUT_TOPI + b * 3 + j] = (float)bi;
    }
}

// gather 3 layers (b128), copy to selected_layers output, weighted sum, partial max over T chunk
// one block covers all of D via float4; grid = (1, B, TCH3)
__global__ void k3_weighted(const float* __restrict__ x, float* __restrict__ ws,
                            float* __restrict__ out) {
    const int tid = threadIdx.x;         // float4 index over D (256*4 = 1024)
    const int b   = blockIdx.y;
    const int tc  = blockIdx.z;
    const int* wsi = (const int*)ws;
    const int i0 = wsi[OFF_TOPI + b * 3 + 0];
    const int i1 = wsi[OFF_TOPI + b * 3 + 1];
    const int i2 = wsi[OFF_TOPI + b * 3 + 2];
    const float v0 = ws[OFF_TOPV + b * 3 + 0];
    const float v1 = ws[OFF_TOPV + b * 3 + 1];
    const float v2 = ws[OFF_TOPV + b * 3 + 2];
    const size_t trow = (size_t)tc * TROW3;
    const float4* x0 = (const float4*)(x + ((size_t)(b * LP1 + i0 + 1) * T_ + trow) * D_) + tid;
    const float4* x1 = (const float4*)(x + ((size_t)(b * LP1 + i1 + 1) * T_ + trow) * D_) + tid;
    const float4* x2 = (const float4*)(x + ((size_t)(b * LP1 + i2 + 1) * T_ + trow) * D_) + tid;
    float4* s0 = (float4*)(out + OUT_SEL + ((size_t)(b * 3 + 0) * T_ + trow) * D_) + tid;
    float4* s1 = (float4*)(out + OUT_SEL + ((size_t)(b * 3 + 1) * T_ + trow) * D_) + tid;
    float4* s2 = (float4*)(out + OUT_SEL + ((size_t)(b * 3 + 2) * T_ + trow) * D_) + tid;
    const int stride = D_ / 4;
    float4 m = make_float4(-3.402823466e38f, -3.402823466e38f,
                           -3.402823466e38f, -3.402823466e38f);
    #pragma unroll 4
    for (int tt = 0; tt < TROW3; ++tt) {
        float4 a0 = x0[tt * stride], a1 = x1[tt * stride], a2 = x2[tt * stride];
        s0[tt * stride] = a0; s1[tt * stride] = a1; s2[tt * stride] = a2;
        m.x = fmaxf(m.x, v0 * a0.x + v1 * a1.x + v2 * a2.x);
        m.y = fmaxf(m.y, v0 * a0.y + v1 * a1.y + v2 * a2.y);
        m.z = fmaxf(m.z, v0 * a0.z + v1 * a1.z + v2 * a2.z);
        m.w = fmaxf(m.w, v0 * a0.w + v1 * a1.w + v2 * a2.w);
    }
    ((float4*)(ws + OFF_PMAX + (size_t)(b * TCH3 + tc) * D_))[tid] = m;
}

// reduce T-chunk maxima, LayerNorm over D, write zero-padded normed (16x1024)
__global__ void k4_ln(const float* __restrict__ gamma, const float* __restrict__ beta,
                      float* __restrict__ ws) {
    __shared__ float rsum[256], rsq[256];
    const int tid = threadIdx.x, b = blockIdx.x;
    const float4* pmax = (const float4*)(ws + OFF_PMAX + (size_t)b * TCH3 * D_);
    float4 m = pmax[tid];
    for (int c = 1; c < TCH3; ++c) {
        float4 v = pmax[c * (D_ / 4) + tid];
        m.x = fmaxf(m.x, v.x); m.y = fmaxf(m.y, v.y);
        m.z = fmaxf(m.z, v.z); m.w = fmaxf(m.w, v.w);
    }
    rsum[tid] = m.x + m.y + m.z + m.w;
    rsq[tid]  = m.x * m.x + m.y * m.y + m.z * m.z + m.w * m.w;
    __syncthreads();
    for (int s = 128; s > 0; s >>= 1) {
        if (tid < s) { rsum[tid] += rsum[tid + s]; rsq[tid] += rsq[tid + s]; }
        __syncthreads();
    }
    const float mu  = rsum[0] * (1.0f / D_);
    const float var = rsq[0] * (1.0f / D_) - mu * mu;
    const float rs  = rsqrtf(var + 1e-5f);
    const float4 g  = ((const float4*)gamma)[tid];
    const float4 be = ((const float4*)beta)[tid];
    float4 o;
    o.x = (m.x - mu) * rs * g.x + be.x;
    o.y = (m.y - mu) * rs * g.y + be.y;
    o.z = (m.z - mu) * rs * g.z + be.z;
    o.w = (m.w - mu) * rs * g.w + be.w;
    ((float4*)(ws + OFF_NORM + b * D_))[tid] = o;
}

// projected = normed(16x1024, rows 8..15 zero) @ w_proj(1024x128) + b_proj
// 8 waves, each owns a 16-wide N slab; 256 K-steps of V_WMMA_F32_16X16X4_F32
__global__ void k5_proj(const float* __restrict__ wproj, const float* __restrict__ bproj,
                        const float* __restrict__ ws, float* __restrict__ out) {
    const int tid  = threadIdx.x;
    const int wave = tid >> 5;
    const int lane = tid & 31;
    const int n0   = wave * 16;
    const int m    = lane & 15;        // A: row M ; B: col N
    const int half = lane >> 4;        // 0: K pair {0,1}, 1: K pair {2,3}
    const int koff = half * 2;
    const float* normed = ws + OFF_NORM;
    v8f acc = {};
    for (int k0 = 0; k0 < D_; k0 += 4) {
        v2f a, bb;
        a.x  = normed[m * D_ + k0 + koff];
        a.y  = normed[m * D_ + k0 + koff + 1];
        bb.x = wproj[(k0 + koff) * P_ + n0 + m];
        bb.y = wproj[(k0 + koff + 1) * P_ + n0 + m];
        acc = __builtin_amdgcn_wmma_f32_16x16x4_f32(false, a, false, bb,
                                                    (short)0, acc, false, false);
    }
    const int n = n0 + m;
    const float bias = bproj[n];
    #pragma unroll
    for (int v = 0; v < 8; ++v) {
        int mm = v + half * 8;
        if (mm < 8) out[mm * P_ + n] = acc[v] + bias;
    }
}

extern "C" void kernel_launch(void* const* d_in, const int* in_sizes, int n_in,
                              void* d_out, int out_size, void* d_ws, size_t ws_size,
                              hipStream_t stream) {
    const float* x        = (const float*)d_in[0];
    const float* w_score  = (const float*)d_in[1];
    const float* b_score  = (const float*)d_in[2];
    const float* ln_gamma = (const float*)d_in[3];
    const float* ln_beta  = (const float*)d_in[4];
    const float* w_proj   = (const float*)d_in[5];
    const float* b_proj   = (const float*)d_in[6];
    float* out = (float*)d_out;
    float* ws  = (float*)d_ws;

    k0_init    <<<32, 256, 0, stream>>>(ws);
    k1_scores  <<<dim3(B_ * L_, TCH1), 256, 0, stream>>>(x, w_score, ws);
    k2_topk    <<<1, 32, 0, stream>>>(b_score, ws, out);
    k3_weighted<<<dim3(1, B_, TCH3), 256, 0, stream>>>(x, ws, out);
    k4_ln      <<<B_, 256, 0, stream>>>(ln_gamma, ln_beta, ws);
    k5_proj    <<<1, 256, 0, stream>>>(w_proj, b_proj, ws, out);
}